// RelationEncoder_52166672777413
// MI455X (gfx1250) — compile-verified
//
#include <hip/hip_runtime.h>
#include <hip/hip_bf16.h>
#include <stdint.h>

typedef __attribute__((ext_vector_type(2))) float v2f;
typedef __attribute__((ext_vector_type(8))) float v8f;
typedef __attribute__((ext_vector_type(4))) int   v4i;

#define LN_EPS 1e-5f

#if defined(__has_builtin)
#if __has_builtin(__builtin_amdgcn_global_load_async_to_lds_b128) && \
    __has_builtin(__builtin_amdgcn_s_wait_asynccnt)
#define USE_ASYNC_LDS 1
#endif
#endif
#ifndef USE_ASYNC_LDS
#define USE_ASYNC_LDS 0
#endif

__device__ __forceinline__ float elu_f(float x) {
    return x > 0.0f ? x : __expf(x) - 1.0f;
}

// ---------------- Kernel 1: split embedding + joint LayerNorm over (N,E) ----
// One block per batch b. a_out layout: [(b*64+n), 72] row-major.
__global__ __launch_bounds__(256) void k_embed_ln(
    const float* __restrict__ x_a, const float* __restrict__ x_b,
    const float* __restrict__ W_a, const float* __restrict__ b_a,
    const float* __restrict__ W_b, const float* __restrict__ b_b,
    const float* __restrict__ ln0_g, const float* __restrict__ ln0_b,
    float* __restrict__ a_out)
{
    __shared__ float pre[4608];          // 64*72
    __shared__ float red[256], red2[256];
    const int b = blockIdx.x, tid = threadIdx.x;
    float lsum = 0.0f, lss = 0.0f;
    for (int idx = tid; idx < 4608; idx += 256) {
        const int n = idx / 72, e = idx - n * 72;
        float acc;
        if (n < 32) {
            acc = b_a[e];
            const float* xr = x_a + (b * 32 + n) * 23;
            #pragma unroll
            for (int d = 0; d < 23; ++d) acc += xr[d] * W_a[d * 72 + e];
        } else {
            acc = b_b[e];
            const float* xr = x_b + (b * 32 + (n - 32)) * 13;
            #pragma unroll
            for (int d = 0; d < 13; ++d) acc += xr[d] * W_b[d * 72 + e];
        }
        pre[idx] = acc;
        lsum += acc; lss += acc * acc;
    }
    red[tid] = lsum; red2[tid] = lss;
    __syncthreads();
    for (int s = 128; s > 0; s >>= 1) {
        if (tid < s) { red[tid] += red[tid + s]; red2[tid] += red2[tid + s]; }
        __syncthreads();
    }
    const float mean = red[0] * (1.0f / 4608.0f);
    const float var  = red2[0] * (1.0f / 4608.0f) - mean * mean;
    const float rs   = rsqrtf(var + LN_EPS);
    for (int idx = tid; idx < 4608; idx += 256) {
        a_out[b * 4608 + idx] = (pre[idx] - mean) * rs * ln0_g[idx] + ln0_b[idx];
    }
}

// ---------------- Kernel 2: u = a @ Wg[0:72] + b_g ; v = a @ Wg[72:144] -----
// f32 WMMA 16x16x4. 16384 tiles (512 m-tiles x 16 n-tiles x {u,v}); 8 waves/block.
__global__ __launch_bounds__(256) void k_uv_wmma(
    const float* __restrict__ a, const float* __restrict__ W_g,
    const float* __restrict__ b_g,
    float* __restrict__ u, float* __restrict__ v)
{
    const int wave = threadIdx.x >> 5;
    const int lane = threadIdx.x & 31;
    const int id    = blockIdx.x * 8 + wave;   // 0..16383
    const int ntile = id & 15;
    const int half  = (id >> 4) & 1;
    const int mtile = id >> 5;                 // 0..511
    const int m0 = mtile * 16, n0 = ntile * 16;
    const int mrow = lane & 15;                // A row / B,C column index
    const int ksub = (lane >> 4) * 2;          // K sub-pair: lanes 16-31 hold K=2,3
    const float* Arow  = a + (m0 + mrow) * 72;
    const float* Wbase = W_g + (half * 72) * 256 + n0 + mrow;
    v8f acc = {};
    #pragma unroll 6
    for (int k = 0; k < 72; k += 4) {
        v2f A, B;
        A.x = Arow[k + ksub];
        A.y = Arow[k + ksub + 1];
        B.x = Wbase[(k + ksub) * 256];
        B.y = Wbase[(k + ksub + 1) * 256];
        acc = __builtin_amdgcn_wmma_f32_16x16x4_f32(
            false, A, false, B, (short)0, acc, false, false);
    }
    float* dst = half ? v : u;
    const float bias = half ? 0.0f : b_g[n0 + mrow];   // fold b_g into u only
    const int rbase = (lane >> 4) * 8;                 // lanes 16-31 hold rows M+8
    #pragma unroll
    for (int r = 0; r < 8; ++r) {
        dst[(m0 + r + rbase) * 256 + n0 + mrow] = acc[r] + bias;
    }
}

// ---------------- Kernel 3 (hot): ELU + LayerNorm(256) over all (i,j) -------
// grid = B*8 blocks of 256 threads (8 waves). Wave w handles i = chunk*8+w,
// loops over j. rows 0-15 of v[b] (16KB) are staged in LDS once per block (async path when
// available), inner loop runs from registers + ds_load_b128.
__global__ __launch_bounds__(256) void k_pair(
    const float* __restrict__ u, const float* __restrict__ v,
    float* __restrict__ partial)
{
    __shared__ float vsh[64 * 256];      // 64 KB: v[b, :, :]
    __shared__ float accsh[8 * 256];     // 8 KB: cross-wave reduction
    const int tid  = threadIdx.x;
    const int wave = tid >> 5, lane = tid & 31;
    const int b = blockIdx.x >> 3, chunk = blockIdx.x & 7;
    const int i = chunk * 8 + wave;

    // Stage 1024 x 16B chunks (4096 floats, rows 0-15 of v[b]) into LDS.
    {
        const float* vglob = v + b * 64 * 256;
#if USE_ASYNC_LDS
        #pragma unroll
        for (int t = tid; t < 1024; t += 256) {
            __builtin_amdgcn_global_load_async_to_lds_b128(
                (v4i*)(vglob + t * 4),
                (v4i*)(vsh + t * 4),
                0, 0);
        }
        __builtin_amdgcn_s_wait_asynccnt(0);
#else
        #pragma unroll
        for (int t = tid; t < 1024; t += 256) {
            ((float4*)vsh)[t] = ((const float4*)vglob)[t];
        }
#endif
    }

    const float4* up = (const float4*)(u + (b * 64 + i) * 256 + lane * 8);
    const float4 u0 = up[0], u1 = up[1];
    __syncthreads();

    float acc[8] = {0,0,0,0,0,0,0,0};
    for (int j = 0; j < 64; ++j) {
        const float4* vp = (const float4*)(vsh + j * 256 + lane * 8);
        const float4 v0 = vp[0], v1 = vp[1];
        float e[8];
        e[0] = elu_f(u0.x + v0.x); e[1] = elu_f(u0.y + v0.y);
        e[2] = elu_f(u0.z + v0.z); e[3] = elu_f(u0.w + v0.w);
        e[4] = elu_f(u1.x + v1.x); e[5] = elu_f(u1.y + v1.y);
        e[6] = elu_f(u1.z + v1.z); e[7] = elu_f(u1.w + v1.w);
        float s = 0.0f, ss = 0.0f;
        #pragma unroll
        for (int k = 0; k < 8; ++k) { s += e[k]; ss += e[k] * e[k]; }
        #pragma unroll
        for (int m = 16; m > 0; m >>= 1) {
            s  += __shfl_xor(s,  m, 32);
            ss += __shfl_xor(ss, m, 32);
        }
        const float mean = s * (1.0f / 256.0f);
        const float var  = ss * (1.0f / 256.0f) - mean * mean;
        const float rs   = rsqrtf(var + LN_EPS);
        #pragma unroll
        for (int k = 0; k < 8; ++k) acc[k] += (e[k] - mean) * rs;
    }
    #pragma unroll
    for (int k = 0; k < 8; ++k) accsh[wave * 256 + lane * 8 + k] = acc[k];
    __syncthreads();
    float t = 0.0f;
    #pragma unroll
    for (int w = 0; w < 8; ++w) t += accsh[w * 256 + tid];
    partial[blockIdx.x * 256 + tid] = t;   // deterministic (no float atomics)
}

// ---------------- Kernel 4a: s[b,h] = lng_g*Σpartial + 4096*lng_b -----------
__global__ __launch_bounds__(256) void k_svec(
    const float* __restrict__ partial, const float* __restrict__ lng_g,
    const float* __restrict__ lng_b, float* __restrict__ s)
{
    const int b = blockIdx.x, h = threadIdx.x;
    float t = 0.0f;
    #pragma unroll
    for (int c = 0; c < 8; ++c) t += partial[(b * 8 + c) * 256 + h];
    s[b * 256 + h] = lng_g[h] * t + 4096.0f * lng_b[h];
}

// ---------------- Kernel 4b: out = elu(s @ W_f + b_f) via f32 WMMA ----------
// M=128, N=128, K=256 -> 8x8 tiles, 8 waves/block, 8 blocks.
__global__ __launch_bounds__(256) void k_out_wmma(
    const float* __restrict__ s, const float* __restrict__ W_f,
    const float* __restrict__ b_f, float* __restrict__ out)
{
    const int wave = threadIdx.x >> 5, lane = threadIdx.x & 31;
    const int id = blockIdx.x * 8 + wave;     // 0..63
    const int mtile = id >> 3, ntile = id & 7;
    const int m0 = mtile * 16, n0 = ntile * 16;
    const int mrow = lane & 15;
    const int ksub = (lane >> 4) * 2;
    const float* Srow = s + (m0 + mrow) * 256;
    const float* Wcol = W_f + n0 + mrow;
    v8f acc = {};
    #pragma unroll 8
    for (int k = 0; k < 256; k += 4) {
        v2f A, B;
        A.x = Srow[k + ksub];
        A.y = Srow[k + ksub + 1];
        B.x = Wcol[(k + ksub) * 128];
        B.y = Wcol[(k + ksub + 1) * 128];
        acc = __builtin_amdgcn_wmma_f32_16x16x4_f32(
            false, A, false, B, (short)0, acc, false, false);
    }
    const float bias = b_f[n0 + mrow];
    const int rbase = (lane >> 4) * 8;
    #pragma unroll
    for (int r = 0; r < 8; ++r) {
        out[(m0 + r + rbase) * 128 + n0 + mrow] = elu_f(acc[r] + bias);
    }
}

extern "C" void kernel_launch(void* const* d_in, const int* in_sizes, int n_in,
                              void* d_out, int out_size, void* d_ws, size_t ws_size,
                              hipStream_t stream) {
    const float* x_a   = (const float*)d_in[0];
    const float* x_b   = (const float*)d_in[1];
    const float* W_a   = (const float*)d_in[2];
    const float* b_a   = (const float*)d_in[3];
    const float* W_b   = (const float*)d_in[4];
    const float* b_b   = (const float*)d_in[5];
    const float* ln0_g = (const float*)d_in[6];
    const float* ln0_b = (const float*)d_in[7];
    const float* W_g   = (const float*)d_in[8];
    const float* b_g   = (const float*)d_in[9];
    const float* lng_g = (const float*)d_in[10];
    const float* lng_b = (const float*)d_in[11];
    const float* W_f   = (const float*)d_in[12];
    const float* b_f   = (const float*)d_in[13];
    float* out = (float*)d_out;

    // Workspace layout (floats): total ~5.08M floats (~20.3 MB)
    float* ws = (float*)d_ws;
    float* a       = ws;                    // 8192*72   = 589824
    float* u       = a + 589824;            // 8192*256  = 2097152
    float* v       = u + 2097152;           // 8192*256  = 2097152
    float* partial = v + 2097152;           // 1024*256  = 262144
    float* svec    = partial + 262144;      // 128*256   = 32768

    k_embed_ln<<<128, 256, 0, stream>>>(x_a, x_b, W_a, b_a, W_b, b_b,
                                        ln0_g, ln0_b, a);
    k_uv_wmma<<<2048, 256, 0, stream>>>(a, W_g, b_g, u, v);
    k_pair<<<1024, 256, 0, stream>>>(u, v, partial);
    k_svec<<<128, 256, 0, stream>>>(partial, lng_g, lng_b, svec);
    k_out_wmma<<<8, 256, 0, stream>>>(svec, W_f, b_f, out);
}